// OriginalBandSplitModule_10660108829568
// MI455X (gfx1250) — compile-verified
//
#include <hip/hip_runtime.h>
#include <hip/hip_bf16.h>

typedef __attribute__((ext_vector_type(2))) float v2f;
typedef __attribute__((ext_vector_type(8))) float v8f;

#define MAX_BANDS  48
#define N_FEAT     128
#define T_DIM      1024
#define F_DIM      1025
#define T_TILE     64
#define MAX_W2     192
#define LDS_STRIDE 193   // odd -> conflict-free for row- and column-order access

struct Band {
  const float* gamma;
  const float* beta;
  const float* W;     // [128, w2] row-major
  const float* bias;  // [128]
  int sb;             // first frequency bin of this band
  int w2;             // 2 * band width
};

struct BandTable { Band b[MAX_BANDS]; };

// One workgroup: (band, batch, 64-wide T tile). 256 threads = 8 waves.
// Wave w owns features [16w, 16w+16) and all four 16-t subtiles.
__global__ __launch_bounds__(256) void bandsplit_ln_gemm(
    const float* __restrict__ xr,
    const float* __restrict__ xi,
    float* __restrict__ out,
    BandTable bt,
    int n_bands)
{
  __shared__ float tile[T_TILE * LDS_STRIDE];
  __shared__ float s_mu[T_TILE];
  __shared__ float s_rs[T_TILE];
  __shared__ float s_ps[4 * T_TILE];   // partial sums
  __shared__ float s_pq[4 * T_TILE];   // partial sum-of-squares

  const int tid  = threadIdx.x;
  const int t0   = blockIdx.x * T_TILE;
  const int bb   = blockIdx.y;
  const int band = blockIdx.z;

  const Band bd = bt.b[band];
  const int w2 = bd.w2;
  if (w2 > MAX_W2) return;              // uniform: safe w.r.t. WMMA EXEC rule
  const int w   = w2 >> 1;
  const int w2p = (w2 + 3) & ~3;        // K padded to multiple of 4

  // GEMM lane mapping (also used to prefetch W early).
  const int lane = tid & 31;
  const int wave = tid >> 5;
  const int half = lane >> 4;           // 0: K pair {k,k+1}; 1: K pair {k+2,k+3}
  const int l16  = lane & 15;
  const int f0   = wave * 16;
  const float* __restrict__ Wrow = bd.W + (size_t)(f0 + l16) * (size_t)w2;

  // Warm this lane's whole W row (<= 768B) into cache while gather/LN run.
  for (int o = 0; o < w2; o += 32)      // 32 floats = 128B line stride
    __builtin_prefetch(Wrow + o, 0, 1);

  // ---- Phase 1: gather [64 t] x [2w] (real | imag), coalesced along T ----
  for (int e = tid; e < T_TILE * w2p; e += 256) {
    const int c  = e >> 6;              // column (input-feature dim)
    const int tt = e & 63;              // t within tile (fastest over tid)
    float v = 0.0f;
    if (c < w2) {
      const int bin = bd.sb + ((c < w) ? c : (c - w));
      const float* src = (c < w) ? xr : xi;
      v = src[((size_t)bb * F_DIM + bin) * T_DIM + (size_t)(t0 + tt)];
    }
    tile[tt * LDS_STRIDE + c] = v;      // pad columns written as EXACT zeros
  }
  __syncthreads();

  // ---- Phase 2: per-t mean / population variance, 4 threads per row ----
  {
    const int r = tid & 63;             // t row
    const int q = tid >> 6;             // partial index 0..3
    float s = 0.0f, s2 = 0.0f;
    for (int c = q; c < w2; c += 4) {
      const float v = tile[r * LDS_STRIDE + c];
      s  += v;
      s2 += v * v;
    }
    s_ps[q * T_TILE + r] = s;
    s_pq[q * T_TILE + r] = s2;
  }
  __syncthreads();
  if (tid < T_TILE) {
    const float s  = s_ps[tid] + s_ps[T_TILE + tid] + s_ps[2 * T_TILE + tid] + s_ps[3 * T_TILE + tid];
    const float s2 = s_pq[tid] + s_pq[T_TILE + tid] + s_pq[2 * T_TILE + tid] + s_pq[3 * T_TILE + tid];
    const float inv = 1.0f / (float)w2;
    const float mu  = s * inv;
    const float var = s2 * inv - mu * mu;
    s_mu[tid] = mu;
    s_rs[tid] = rsqrtf(var + 1e-5f);
  }
  __syncthreads();

  // ---- Phase 3: normalize in place (pad columns stay exactly zero) ----
  for (int e = tid; e < T_TILE * w2; e += 256) {
    const int c  = e >> 6;
    const int tt = e & 63;
    const float v = tile[tt * LDS_STRIDE + c];
    tile[tt * LDS_STRIDE + c] = (v - s_mu[tt]) * s_rs[tt] * bd.gamma[c] + bd.beta[c];
  }
  __syncthreads();

  // ---- Phase 4: fp32 WMMA GEMM. A = W rows (M=feature), B = normed^T (N=t) ----
  // A-fragment OOB K slots need no masking: the matching B rows are LDS pad
  // columns (exact zeros), so any FINITE A value contributes 0. We therefore
  // just clamp the offset (one v_cndmask) and load unconditionally, which lets
  // the prefetched value stay untouched until its consuming WMMA next
  // iteration -> the s_wait_loadcnt sinks past 4 DS loads + 4 WMMAs.
  v8f acc0 = {}, acc1 = {}, acc2 = {}, acc3 = {};

  {
    const int kh = half * 2;
    v2f a;                               // first iteration always in range
    a.x = Wrow[kh];
    a.y = Wrow[kh + 1];

    for (int k = 0; k < w2p; k += 4) {
      // Unconditional clamped prefetch of A for iteration k+4 (straight-line).
      const int krn_raw = k + 4 + kh;
      const int krn = (krn_raw < w2) ? krn_raw : 0;    // per-lane select only
      v2f an;
      an.x = Wrow[krn];
      an.y = Wrow[krn + 1];

      const int krow = k + kh;
      const float* bl = tile + krow;                    // pad cols are 0 in LDS
      v2f b0, b1, b2, b3;
      b0.x = bl[(l16 +  0) * LDS_STRIDE]; b0.y = bl[(l16 +  0) * LDS_STRIDE + 1];
      b1.x = bl[(l16 + 16) * LDS_STRIDE]; b1.y = bl[(l16 + 16) * LDS_STRIDE + 1];
      b2.x = bl[(l16 + 32) * LDS_STRIDE]; b2.y = bl[(l16 + 32) * LDS_STRIDE + 1];
      b3.x = bl[(l16 + 48) * LDS_STRIDE]; b3.y = bl[(l16 + 48) * LDS_STRIDE + 1];

      acc0 = __builtin_amdgcn_wmma_f32_16x16x4_f32(false, a, false, b0, (short)0, acc0, false, false);
      acc1 = __builtin_amdgcn_wmma_f32_16x16x4_f32(false, a, false, b1, (short)0, acc1, false, false);
      acc2 = __builtin_amdgcn_wmma_f32_16x16x4_f32(false, a, false, b2, (short)0, acc2, false, false);
      acc3 = __builtin_amdgcn_wmma_f32_16x16x4_f32(false, a, false, b3, (short)0, acc3, false, false);

      a = an;
    }
  }

  // ---- Phase 5: bias + store. D vgpr j: lanes0-15 -> f0+j, lanes16-31 -> f0+8+j;
  //      N = l16 -> t, so each half-wave writes a contiguous 64B segment. ----
  const size_t fstride = (size_t)n_bands * (size_t)T_DIM;   // between feature rows
  const int    fbase   = f0 + half * 8;
  float* op = out + (((size_t)bb * N_FEAT + fbase) * (size_t)n_bands + band) * T_DIM
                  + (size_t)(t0 + l16);
  #pragma unroll
  for (int j = 0; j < 8; ++j) {
    const float bv = bd.bias[fbase + j];
    op[ 0] = acc0[j] + bv;
    op[16] = acc1[j] + bv;
    op[32] = acc2[j] + bv;
    op[48] = acc3[j] + bv;
    op += fstride;
  }
}

extern "C" void kernel_launch(void* const* d_in, const int* in_sizes, int n_in,
                              void* d_out, int out_size, void* d_ws, size_t ws_size,
                              hipStream_t stream) {
  (void)d_ws; (void)ws_size; (void)out_size;
  const float* xr = (const float*)d_in[0];
  const float* xi = (const float*)d_in[1];
  // d_in[2] = sample_rate scalar (band ranges recovered from param sizes instead)
  float* out = (float*)d_out;

  int n_bands = (n_in - 3) / 4;
  if (n_bands > MAX_BANDS) n_bands = MAX_BANDS;

  BandTable bt;
  int sb = 0;                                  // bands are contiguous in bins
  for (int j = 0; j < n_bands; ++j) {
    const int base = 3 + 4 * j;                // gamma, beta, W, b per band
    bt.b[j].gamma = (const float*)d_in[base + 0];
    bt.b[j].beta  = (const float*)d_in[base + 1];
    bt.b[j].W     = (const float*)d_in[base + 2];
    bt.b[j].bias  = (const float*)d_in[base + 3];
    const int w2  = in_sizes[base + 0];        // |gamma| = 2*(eb-sb)
    bt.b[j].sb = sb;
    bt.b[j].w2 = w2;
    sb += w2 >> 1;
  }

  const int Bsz = in_sizes[0] / (F_DIM * T_DIM);
  dim3 grid(T_DIM / T_TILE, Bsz, n_bands);
  bandsplit_ln_gemm<<<grid, 256, 0, stream>>>(xr, xi, out, bt, n_bands);
}